// GCN_17300128268933
// MI455X (gfx1250) — compile-verified
//
#include <hip/hip_runtime.h>

#define N_NODES   100000
#define N_EDGES   1600000
#define IN_FEAT   512
#define HIDDEN    32
#define N_CLASSES 16

typedef float v2f __attribute__((ext_vector_type(2)));
typedef float v8f __attribute__((ext_vector_type(8)));

#if defined(__gfx1250__) && __has_builtin(__builtin_amdgcn_global_load_async_to_lds_b128)
#define HAVE_ASYNC_LDS 1
// param type per hipcc diagnostic: "int __vector(4) __device__ *"
typedef int v4i_b128 __attribute__((vector_size(16)));
#else
#define HAVE_ASYNC_LDS 0
#endif

// ---------------------------------------------------------------- utilities

__global__ void zero_f32(float* __restrict__ p, int n) {
  int i = blockIdx.x * blockDim.x + threadIdx.x;
  int stride = gridDim.x * blockDim.x;
  for (; i < n; i += stride) p[i] = 0.0f;
}

__global__ void degree_kernel(const int* __restrict__ col, float* __restrict__ deg,
                              int n_edges) {
  int e = blockIdx.x * blockDim.x + threadIdx.x;
  if (e < n_edges) atomicAdd(&deg[col[e]], 1.0f);
}

// dis[i] = rsqrt(deg_edges[i] + 1)   (every node gets exactly one self-loop)
__global__ void rsqrt_kernel(float* __restrict__ deg, int n) {
  int i = blockIdx.x * blockDim.x + threadIdx.x;
  if (i < n) deg[i] = rsqrtf(deg[i] + 1.0f);
}

// out[c * rows + r] = in[r * cols + c]
__global__ void transpose_kernel(const float* __restrict__ in, float* __restrict__ out,
                                 int rows, int cols) {
  int i = blockIdx.x * blockDim.x + threadIdx.x;
  if (i < rows * cols) {
    const int r = i / cols;
    const int c = i - r * cols;
    out[c * rows + r] = in[i];
  }
}

// ---------------------------------------------------------------- GEMM 1
// h_lin[100000 x 32] = x[100000 x 512] @ W1[512 x 32]  (W1 given transposed:
// W1t[32 x 512]).  Block: 64 threads (2 waves); wave w computes the 16x16
// tile at rows [16*blockIdx.x,+16), cols [16*w,+16).  X tile staged in LDS.
#define XS_STRIDE 516  // 512 + 4 pad -> conflict-free column-broadcast reads

__global__ __launch_bounds__(64) void gemm1_wmma(const float* __restrict__ x,
                                                 const float* __restrict__ W1t,
                                                 float* __restrict__ h_lin) {
  __shared__ float Xs[16 * XS_STRIDE];
  const int row0 = blockIdx.x * 16;
  const int tid  = threadIdx.x;

  // Stage 16 x 512 floats (2048 float4 slots) with 64 threads.
#if HAVE_ASYNC_LDS
  for (int i = tid; i < 16 * 128; i += 64) {
    const int r  = i >> 7;
    const int c4 = i & 127;
    __builtin_amdgcn_global_load_async_to_lds_b128(
        (__attribute__((address_space(1))) v4i_b128*)(x + (size_t)(row0 + r) * IN_FEAT + c4 * 4),
        (__attribute__((address_space(3))) v4i_b128*)&Xs[r * XS_STRIDE + c4 * 4],
        0, 0);
  }
  asm volatile("s_wait_asynccnt 0x0" ::: "memory");
#else
  for (int i = tid; i < 16 * 128; i += 64) {
    const int r  = i >> 7;
    const int c4 = i & 127;
    const float4 v =
        *reinterpret_cast<const float4*>(x + (size_t)(row0 + r) * IN_FEAT + c4 * 4);
    *reinterpret_cast<float4*>(&Xs[r * XS_STRIDE + c4 * 4]) = v;
  }
#endif
  __syncthreads();

  const int lane  = tid & 31;
  const int ntile = tid >> 5;                 // 0 or 1
  const int m     = lane & 15;                // A row / output N-lane
  const int kb    = (lane >> 4) << 1;         // K sub-offset: 0 or 2
  const int col   = ntile * 16 + (lane & 15); // output column

  v8f c = {};
  for (int k = 0; k < IN_FEAT; k += 4) {
    const int k2 = k + kb;
    v2f a, b;
    a.x = Xs[m * XS_STRIDE + k2];
    a.y = Xs[m * XS_STRIDE + k2 + 1];
    const float2 bw = *reinterpret_cast<const float2*>(&W1t[(size_t)col * IN_FEAT + k2]);
    b.x = bw.x;
    b.y = bw.y;
    c = __builtin_amdgcn_wmma_f32_16x16x4_f32(false, a, false, b,
                                              (short)0, c, false, false);
  }

  const int mh = (lane >> 4) << 3;  // M offset of this half-wave's C rows
#pragma unroll
  for (int i = 0; i < 8; ++i) {
    h_lin[(size_t)(row0 + mh + i) * HIDDEN + col] = c[i];
  }
}

// ---------------------------------------------------------------- GEMM 2
// h2_lin[100000 x 16] = relu(h_agg + b1)[100000 x 32] @ W2[32 x 16]
// (W2 given transposed: W2t[16 x 32]).  Bias+ReLU fused into the A load —
// each h_agg element is consumed exactly once here.  One wave per 16 rows.
__global__ __launch_bounds__(32) void gemm2_wmma(const float* __restrict__ h_agg,
                                                 const float* __restrict__ b1,
                                                 const float* __restrict__ W2t,
                                                 float* __restrict__ h2_lin) {
  const int row0 = blockIdx.x * 16;
  const int lane = threadIdx.x & 31;
  const int m    = lane & 15;
  const int kb   = (lane >> 4) << 1;
  const int col  = lane & 15;

  v8f c = {};
#pragma unroll
  for (int k = 0; k < HIDDEN; k += 4) {
    const int k2 = k + kb;
    const float2 av = *reinterpret_cast<const float2*>(&h_agg[(size_t)(row0 + m) * HIDDEN + k2]);
    const float2 bb = *reinterpret_cast<const float2*>(&b1[k2]);
    const float2 bw = *reinterpret_cast<const float2*>(&W2t[(size_t)col * HIDDEN + k2]);
    v2f a, b;
    a.x = fmaxf(av.x + bb.x, 0.0f);
    a.y = fmaxf(av.y + bb.y, 0.0f);
    b.x = bw.x;
    b.y = bw.y;
    c = __builtin_amdgcn_wmma_f32_16x16x4_f32(false, a, false, b,
                                              (short)0, c, false, false);
  }

  const int mh = (lane >> 4) << 3;
#pragma unroll
  for (int i = 0; i < 8; ++i) {
    h2_lin[(size_t)(row0 + mh + i) * N_CLASSES + col] = c[i];
  }
}

// ---------------------------------------------------------------- scatter
// One F-lane group per edge: dst[col] += src[row] * dis[row]*dis[col].
// Edges [n_edges, n_edges+n_nodes) are the implicit self-loops (w = dis^2).
template <int F>
__global__ void scatter_kernel(const float* __restrict__ src, float* __restrict__ dst,
                               const int* __restrict__ erow, const int* __restrict__ ecol,
                               const float* __restrict__ dis, int n_edges, int n_nodes) {
  const int per_block = 256 / F;
  int eidx = blockIdx.x * per_block + (int)(threadIdx.x / F);
  if (F == 32) {
    // edge index is wave-uniform for 32 lanes/edge -> make it explicit so the
    // index/norm loads can be scalarized
    eidx = __builtin_amdgcn_readfirstlane(eidx);
  }
  const int f = (int)(threadIdx.x % F);
  const int total = n_edges + n_nodes;
  if (eidx >= total) return;

  int r, c;
  float w;
  if (eidx < n_edges) {
    r = erow[eidx];
    c = ecol[eidx];
    w = dis[r] * dis[c];
  } else {
    r = c = eidx - n_edges;
    const float d = dis[r];
    w = d * d;
  }
  const float v = src[(size_t)r * F + f] * w;
  atomicAdd(&dst[(size_t)c * F + f], v);
}

// ---------------------------------------------------------------- epilogue

__global__ void logsoftmax_kernel(const float* __restrict__ h,
                                  const float* __restrict__ b2,
                                  float* __restrict__ out, int n_nodes) {
  int node = blockIdx.x * blockDim.x + threadIdx.x;
  if (node >= n_nodes) return;
  float v[N_CLASSES];
  float mx = -3.402823466e38f;
#pragma unroll
  for (int j = 0; j < N_CLASSES; ++j) {
    v[j] = h[(size_t)node * N_CLASSES + j] + b2[j];
    mx = fmaxf(mx, v[j]);
  }
  float s = 0.0f;
#pragma unroll
  for (int j = 0; j < N_CLASSES; ++j) s += expf(v[j] - mx);
  const float l = logf(s);
#pragma unroll
  for (int j = 0; j < N_CLASSES; ++j)
    out[(size_t)node * N_CLASSES + j] = v[j] - mx - l;
}

// ---------------------------------------------------------------- launcher

extern "C" void kernel_launch(void* const* d_in, const int* in_sizes, int n_in,
                              void* d_out, int out_size, void* d_ws, size_t ws_size,
                              hipStream_t stream) {
  (void)in_sizes; (void)n_in; (void)out_size; (void)ws_size;

  const float* x  = (const float*)d_in[0];
  const int*   ei = (const int*)d_in[1];
  const float* W1 = (const float*)d_in[2];
  const float* b1 = (const float*)d_in[3];
  const float* W2 = (const float*)d_in[4];
  const float* b2 = (const float*)d_in[5];
  float*       out = (float*)d_out;

  const int* erow = ei;
  const int* ecol = ei + N_EDGES;

  // ws layout (floats): [zeroed: dis | h_agg | h2_agg] | h_lin | h2_lin | W1t | W2t
  float* ws     = (float*)d_ws;
  float* dis    = ws;                                     // N
  float* h_agg  = dis + N_NODES;                          // N*32
  float* h2_agg = h_agg + (size_t)N_NODES * HIDDEN;       // N*16
  float* h_lin  = h2_agg + (size_t)N_NODES * N_CLASSES;   // N*32
  float* h2_lin = h_lin + (size_t)N_NODES * HIDDEN;       // N*16
  float* W1t    = h2_lin + (size_t)N_NODES * N_CLASSES;   // 512*32
  float* W2t    = W1t + IN_FEAT * HIDDEN;                 // 32*16

  const int zero_f32_count = N_NODES * (1 + HIDDEN + N_CLASSES);  // 4.9M
  zero_f32<<<1024, 256, 0, stream>>>(ws, zero_f32_count);

  transpose_kernel<<<(IN_FEAT * HIDDEN + 255) / 256, 256, 0, stream>>>(W1, W1t,
                                                                       IN_FEAT, HIDDEN);
  transpose_kernel<<<(HIDDEN * N_CLASSES + 255) / 256, 256, 0, stream>>>(W2, W2t,
                                                                         HIDDEN, N_CLASSES);

  degree_kernel<<<(N_EDGES + 255) / 256, 256, 0, stream>>>(ecol, dis, N_EDGES);
  rsqrt_kernel<<<(N_NODES + 255) / 256, 256, 0, stream>>>(dis, N_NODES);

  // Layer 1
  gemm1_wmma<<<N_NODES / 16, 64, 0, stream>>>(x, W1t, h_lin);
  {
    const int total = N_EDGES + N_NODES;
    const int per_block = 256 / 32;
    scatter_kernel<HIDDEN><<<(total + per_block - 1) / per_block, 256, 0, stream>>>(
        h_lin, h_agg, erow, ecol, dis, N_EDGES, N_NODES);
  }

  // Layer 2 (bias+ReLU fused into the A-fragment load)
  gemm2_wmma<<<N_NODES / 16, 32, 0, stream>>>(h_agg, b1, W2t, h2_lin);
  {
    const int total = N_EDGES + N_NODES;
    const int per_block = 256 / 16;
    scatter_kernel<N_CLASSES><<<(total + per_block - 1) / per_block, 256, 0, stream>>>(
        h2_lin, h2_agg, erow, ecol, dis, N_EDGES, N_NODES);
  }

  logsoftmax_kernel<<<(N_NODES + 255) / 256, 256, 0, stream>>>(h2_agg, b2, out, N_NODES);
}